// ObjectiveFunction_14542759264809
// MI455X (gfx1250) — compile-verified
//
#include <hip/hip_runtime.h>
#include <hip/hip_bf16.h>
#include <math.h>

// ---------------------------------------------------------------------------
// Problem constants (match reference)
// ---------------------------------------------------------------------------
#define NUM_VERTS 4000000
#define NUM_FACES 1000000
#define NB_CLOSE  2048   // blocks for closeness kernel
#define NB_FACE   1024   // blocks for face kernel
#define BLK       256    // 8 waves of 32

typedef float v2f __attribute__((ext_vector_type(2)));
typedef float v4f __attribute__((ext_vector_type(4)));
typedef float v8f __attribute__((ext_vector_type(8)));
typedef int   v4i __attribute__((ext_vector_type(4)));

// ---------------------------------------------------------------------------
// Wave32 reduction via V_WMMA_F32_16X16X4_F32.
// A (16x4 f32) = 2 VGPRs/lane -> exactly the 64 values {a0,a1} x 32 lanes.
// B = all-ones => D[m][n] = rowsum(m) for every n.
// Lane n (n<16) holds D[0..7][n], lane n+16 holds D[8..15][n]; summing the
// 8 components and adding the xor-16 partner gives sum over all lanes of
// (a0+a1) in every lane.  EXEC must be all ones at the call site.
// ---------------------------------------------------------------------------
__device__ __forceinline__ float wave_reduce_wmma(float a0, float a1) {
  v2f A; A[0] = a0;   A[1] = a1;
  v2f B; B[0] = 1.0f; B[1] = 1.0f;
  v8f C = {};
  v8f D = __builtin_amdgcn_wmma_f32_16x16x4_f32(
      /*neg_a=*/false, A, /*neg_b=*/false, B,
      /*c_mod=*/(short)0, C, /*reuse_a=*/false, /*reuse_b=*/false);
  float s = 0.0f;
#pragma unroll
  for (int k = 0; k < 8; ++k) s += D[k];
  // SWAPX16: ds_swizzle group-of-32, xor_mask=0x10, and_mask=0x1f
  int sw = __builtin_amdgcn_ds_swizzle(__builtin_bit_cast(int, s), 0x401F);
  s += __builtin_bit_cast(float, sw);
  return s;
}

// ---------------------------------------------------------------------------
// Kernel 1: closeness = sum_i vw[i] * (dx+dy+dz)^2
// 4 vertices per thread-iteration: 3x b128 (w) + 3x b128 NT (p) + 1x b128 NT (vw)
// ---------------------------------------------------------------------------
__global__ void __launch_bounds__(BLK)
k_closeness(const v4f* __restrict__ w4, const v4f* __restrict__ p4,
            const v4f* __restrict__ vw4, float* __restrict__ partial) {
  const int tid    = blockIdx.x * blockDim.x + threadIdx.x;
  const int stride = gridDim.x * blockDim.x;
  const int ngroups = NUM_VERTS / 4;   // 1,000,000 exactly

  float acc = 0.0f;
  for (int g = tid; g < ngroups; g += stride) {
    const v4f a0 = w4[3 * g + 0];
    const v4f a1 = w4[3 * g + 1];
    const v4f a2 = w4[3 * g + 2];
    const v4f b0 = __builtin_nontemporal_load(&p4[3 * g + 0]);
    const v4f b1 = __builtin_nontemporal_load(&p4[3 * g + 1]);
    const v4f b2 = __builtin_nontemporal_load(&p4[3 * g + 2]);
    const v4f vw = __builtin_nontemporal_load(&vw4[g]);

    const float t0 = (a0.x - b0.x) + (a0.y - b0.y) + (a0.z - b0.z);
    const float t1 = (a0.w - b0.w) + (a1.x - b1.x) + (a1.y - b1.y);
    const float t2 = (a1.z - b1.z) + (a1.w - b1.w) + (a2.x - b2.x);
    const float t3 = (a2.y - b2.y) + (a2.z - b2.z) + (a2.w - b2.w);

    acc = fmaf(vw.x * t0, t0, acc);
    acc = fmaf(vw.y * t1, t1, acc);
    acc = fmaf(vw.z * t2, t2, acc);
    acc = fmaf(vw.w * t3, t3, acc);
  }

  // full-wave WMMA reduction (uniform call site, EXEC all ones)
  const float wsum = wave_reduce_wmma(acc, 0.0f);

  __shared__ float red[BLK / 32];
  const int lane = threadIdx.x & 31;
  const int wv   = threadIdx.x >> 5;
  if (lane == 0) red[wv] = wsum;
  __syncthreads();
  if (threadIdx.x == 0) {
    float t = 0.0f;
#pragma unroll
    for (int k = 0; k < BLK / 32; ++k) t += red[k];
    partial[blockIdx.x] = t;
  }
}

// ---------------------------------------------------------------------------
// Kernel 2: per-face angle + determinant terms (L2 gathers of weights)
// ---------------------------------------------------------------------------
__global__ void __launch_bounds__(BLK)
k_face(const float* __restrict__ w, const v4i* __restrict__ faces,
       float* __restrict__ partial) {
  const int tid    = blockIdx.x * blockDim.x + threadIdx.x;
  const int stride = gridDim.x * blockDim.x;

  float acc = 0.0f;
  for (int f = tid; f < NUM_FACES; f += stride) {
    const v4i q = __builtin_nontemporal_load(&faces[f]);  // read-once stream
    int id[4] = {q.x, q.y, q.z, q.w};

    float px[4], py[4], pz[4];
#pragma unroll
    for (int j = 0; j < 4; ++j) {
      const float* vp = w + (size_t)id[j] * 3;
      px[j] = vp[0]; py[j] = vp[1]; pz[j] = vp[2];
    }

    float bnx[4], bny[4], bnz[4];   // normalized "bc" edge vectors
    float angsum = 0.0f;
#pragma unroll
    for (int j = 0; j < 4; ++j) {
      const int jn = (j + 1) & 3;   // roll -1 : next vertex
      const int jp = (j + 3) & 3;   // roll +1 : previous vertex
      float bax = px[jp] - px[j], bay = py[jp] - py[j], baz = pz[jp] - pz[j];
      float bcx = px[jn] - px[j], bcy = py[jn] - py[j], bcz = pz[jn] - pz[j];
      const float ra = 1.0f / sqrtf(bax * bax + bay * bay + baz * baz);
      const float rc = 1.0f / sqrtf(bcx * bcx + bcy * bcy + bcz * bcz);
      bax *= ra; bay *= ra; baz *= ra;
      bcx *= rc; bcy *= rc; bcz *= rc;
      bnx[j] = bcx; bny[j] = bcy; bnz[j] = bcz;
      float c = bax * bcx + bay * bcy + baz * bcz;
      c = fminf(fmaxf(c, -1.0f + 1e-6f), 1.0f - 1e-6f);
      angsum += acosf(c);
    }
    const float plan = angsum - 6.28318530717958647692f;  // (n-2)*pi, n=4
    acc = fmaf(plan, plan, acc);

    // TRIPLES = {0,1,2},{0,1,3},{0,2,3},{1,2,3}
    const int T0[4] = {0, 0, 0, 1};
    const int T1[4] = {1, 1, 2, 2};
    const int T2[4] = {2, 3, 3, 3};
#pragma unroll
    for (int t = 0; t < 4; ++t) {
      const int a = T0[t], b = T1[t], c2 = T2[t];
      const float cx = bny[b] * bnz[c2] - bnz[b] * bny[c2];
      const float cy = bnz[b] * bnx[c2] - bnx[b] * bnz[c2];
      const float cz = bnx[b] * bny[c2] - bny[b] * bnx[c2];
      const float det = bnx[a] * cx + bny[a] * cy + bnz[a] * cz;
      acc = fmaf(det, det, acc);
    }
  }

  const float wsum = wave_reduce_wmma(acc, 0.0f);

  __shared__ float red[BLK / 32];
  const int lane = threadIdx.x & 31;
  const int wv   = threadIdx.x >> 5;
  if (lane == 0) red[wv] = wsum;
  __syncthreads();
  if (threadIdx.x == 0) {
    float t = 0.0f;
#pragma unroll
    for (int k = 0; k < BLK / 32; ++k) t += red[k];
    partial[blockIdx.x] = t;
  }
}

// ---------------------------------------------------------------------------
// Kernel 3: deterministic final reduction + scaling
// energy = closeness/SURFACE_AREA^2 + (angle+det)/F
// ---------------------------------------------------------------------------
__global__ void __launch_bounds__(BLK)
k_finalize(const float* __restrict__ ws, float* __restrict__ out) {
  __shared__ float smc[BLK];
  __shared__ float smf[BLK];
  const int t = threadIdx.x;
  float c = 0.0f, f = 0.0f;
  for (int i = t; i < NB_CLOSE; i += BLK) c += ws[i];
  for (int i = t; i < NB_FACE; i += BLK) f += ws[NB_CLOSE + i];
  smc[t] = c; smf[t] = f;
  __syncthreads();
  for (int s = BLK / 2; s > 0; s >>= 1) {
    if (t < s) { smc[t] += smc[t + s]; smf[t] += smf[t + s]; }
    __syncthreads();
  }
  if (t == 0) out[0] = smc[0] / 10000.0f + smf[0] / 1000000.0f;
}

// ---------------------------------------------------------------------------
// Launch: inputs are (weights, initial_positions, vertex_weights, faces)
// ---------------------------------------------------------------------------
extern "C" void kernel_launch(void* const* d_in, const int* in_sizes, int n_in,
                              void* d_out, int out_size, void* d_ws, size_t ws_size,
                              hipStream_t stream) {
  const v4f* w4    = (const v4f*)d_in[0];
  const v4f* p4    = (const v4f*)d_in[1];
  const v4f* vw4   = (const v4f*)d_in[2];
  const float* w   = (const float*)d_in[0];
  const v4i* faces = (const v4i*)d_in[3];
  float* ws  = (float*)d_ws;   // [0,NB_CLOSE) closeness partials, then NB_FACE face partials
  float* out = (float*)d_out;

  k_closeness<<<NB_CLOSE, BLK, 0, stream>>>(w4, p4, vw4, ws);
  k_face<<<NB_FACE, BLK, 0, stream>>>(w, faces, ws + NB_CLOSE);
  k_finalize<<<1, BLK, 0, stream>>>(ws, out);
}